// GPT2Block_15556371546458
// MI455X (gfx1250) — compile-verified
//
#include <hip/hip_runtime.h>
#include <math.h>

// ---------------------------------------------------------------- constants
#define H_DIM   2048
#define N_HEADS 16
#define HEAD_D  128
#define D_FF    8192
#define B_SZ    32
#define BLK_T   64
#define MAXBLK  32
#define NUMBLK  1024

typedef float v2f __attribute__((ext_vector_type(2)));
typedef float v4f __attribute__((ext_vector_type(4)));
typedef float v8f __attribute__((ext_vector_type(8)));

// ---------------------------------------------------------------- LayerNorm
// One block per row (2048 elems, 256 threads * 8 each), two-stage LDS reduce.
__global__ __launch_bounds__(256) void ln_kernel(const float* __restrict__ x,
                                                 const float* __restrict__ g,
                                                 const float* __restrict__ b,
                                                 float* __restrict__ out) {
    const int row = blockIdx.x;
    const int t   = threadIdx.x;
    const float* xr = x + (size_t)row * H_DIM;
    float v[8];
    float s = 0.f;
#pragma unroll
    for (int i = 0; i < 8; ++i) { v[i] = xr[t + 256 * i]; s += v[i]; }

    __shared__ float red[256];
    red[t] = s; __syncthreads();
    for (int o = 128; o > 0; o >>= 1) { if (t < o) red[t] += red[t + o]; __syncthreads(); }
    const float mu = red[0] * (1.0f / H_DIM);
    __syncthreads();

    float vs = 0.f;
#pragma unroll
    for (int i = 0; i < 8; ++i) { const float d = v[i] - mu; vs += d * d; }
    red[t] = vs; __syncthreads();
    for (int o = 128; o > 0; o >>= 1) { if (t < o) red[t] += red[t + o]; __syncthreads(); }
    const float rstd = rsqrtf(red[0] * (1.0f / H_DIM) + 1e-5f);

#pragma unroll
    for (int i = 0; i < 8; ++i) {
        const int c = t + 256 * i;
        out[(size_t)row * H_DIM + c] = (v[i] - mu) * rstd * g[c] + b[c];
    }
}

// ---------------------------------------------------------------- WMMA GEMM
// C[32,N] = A[32,K] @ B[K,N] + bias (+res) (ACT==1 -> exact-erf GELU).
// N, K compile-time -> B accesses fold into constant immediate offsets.
// Weights (B) are streamed once -> non-temporal loads keep L2 free for the
// small activation matrix A, which every N-tile workgroup re-reads.
// 256 threads = 8 waves; each wave takes K/8 and produces two 16x16 f32
// WMMA tiles; partials merge through LDS in fixed wave order (deterministic).
template <int N, int K, int ACT, bool RES>
__global__ __launch_bounds__(256) void gemm32(const float* __restrict__ A,
                                              const float* __restrict__ Bm,
                                              const float* __restrict__ bias,
                                              const float* __restrict__ res,
                                              float* __restrict__ out) {
    __shared__ float lds[8 * 512];

    const int lane = threadIdx.x & 31;
    const int wave = threadIdx.x >> 5;
    const int n0   = blockIdx.x * 16;
    const int col  = n0 + (lane & 15);
    const int koff = (lane >> 4) * 2;        // ISA A/B fragment K-pair select
    const int rA0  = (lane & 15);            // M tile 0 rows 0..15
    const int rA1  = rA0 + 16;               // M tile 1 rows 16..31

    constexpr int KCHUNK = K >> 3;           // K per wave (multiple of 16)
    const int kbeg = wave * KCHUNK;

    const float* aptr0 = A  + (size_t)rA0 * K + kbeg + koff;
    const float* aptr1 = A  + (size_t)rA1 * K + kbeg + koff;
    const float* bptr  = Bm + (size_t)(kbeg + koff) * N + col;

    v8f c0 = {}; v8f c1 = {};
    for (int k0 = 0; k0 < KCHUNK; k0 += 16) {
        // batch all 16 loads (4 k-steps), then 8 WMMAs -> deep load clause
        v2f a0[4], a1[4], bb[4];
#pragma unroll
        for (int u = 0; u < 4; ++u) {
            a0[u] = *(const v2f*)(aptr0 + 4 * u);
            a1[u] = *(const v2f*)(aptr1 + 4 * u);
            bb[u].x = __builtin_nontemporal_load(bptr + (4 * u)     * N);
            bb[u].y = __builtin_nontemporal_load(bptr + (4 * u + 1) * N);
        }
#pragma unroll
        for (int u = 0; u < 4; ++u) {
            c0 = __builtin_amdgcn_wmma_f32_16x16x4_f32(false, a0[u], false, bb[u], (short)0, c0, false, false);
            c1 = __builtin_amdgcn_wmma_f32_16x16x4_f32(false, a1[u], false, bb[u], (short)0, c1, false, false);
        }
        aptr0 += 16; aptr1 += 16; bptr += 16 * N;
    }

    // scatter partial tiles to LDS at their logical (row,col) position
    const int colL  = lane & 15;
    const int rbase = 8 * (lane >> 4);       // C/D layout: lanes 16-31 -> M+8
#pragma unroll
    for (int j = 0; j < 8; ++j) {
        lds[wave * 512 + (rbase + j)      * 16 + colL] = c0[j];
        lds[wave * 512 + (rbase + j + 16) * 16 + colL] = c1[j];
    }
    __syncthreads();

    // fixed-order cross-wave sum + epilogue (512 outputs, 2 per thread)
    for (int e = threadIdx.x; e < 512; e += 256) {
        const int row  = e >> 4;
        const int ocol = n0 + (e & 15);
        float acc = 0.f;
#pragma unroll
        for (int w = 0; w < 8; ++w) acc += lds[w * 512 + e];
        acc += bias[ocol];
        if (ACT == 1) acc = 0.5f * acc * (1.0f + erff(acc * 0.70710678118654752f));
        if (RES) acc += res[(size_t)row * N + ocol];
        out[(size_t)row * N + ocol] = acc;
    }
}

// ---------------------------------------------------------------- attention
// One workgroup per (batch, head). 8 waves; lane l owns head dims [4l,4l+4).
// Two positions per iteration (stride 16) to double KV loads in flight on
// the dominant ~1 GB stream; KV loads are non-temporal (streamed once).
// Online softmax per wave, deterministic LDS merge across waves.
__global__ __launch_bounds__(256) void attn_kernel(const float* __restrict__ qkv,
                                                   const float* __restrict__ kv,
                                                   const int*   __restrict__ block_tables,
                                                   const int*   __restrict__ seq_lens,
                                                   float* __restrict__ out) {
    const int b    = blockIdx.x >> 4;
    const int h    = blockIdx.x & 15;
    const int lane = threadIdx.x & 31;
    const int wave = threadIdx.x >> 5;

    const float* Kbase = kv;
    const float* Vbase = kv + (size_t)NUMBLK * BLK_T * N_HEADS * HEAD_D;
    const int*   bt    = block_tables + b * MAXBLK;
    const int    L     = seq_lens[b];

    const v4f q = *(const v4f*)(qkv + (size_t)b * 3 * H_DIM + h * HEAD_D + lane * 4);
    const float scale = 0.088388347648318447f;   // 1/sqrt(128)

    float m = -1e30f, l = 0.f;
    v4f acc = {0.f, 0.f, 0.f, 0.f};

    for (int s = wave; s < L; s += 16) {
        const int s2   = s + 8;
        const bool has2 = (s2 < L);

        const int blk1 = bt[s >> 6];
        const size_t b1 = (((size_t)blk1 * BLK_T + (s & 63)) * N_HEADS + h) * HEAD_D + lane * 4;
        size_t b2 = b1;
        if (has2) {
            const int blk2 = bt[s2 >> 6];
            b2 = (((size_t)blk2 * BLK_T + (s2 & 63)) * N_HEADS + h) * HEAD_D + lane * 4;
        }

        const v4f k1 = __builtin_nontemporal_load((const v4f*)(Kbase + b1));
        v4f k2 = {0.f, 0.f, 0.f, 0.f};
        if (has2) k2 = __builtin_nontemporal_load((const v4f*)(Kbase + b2));

        // prefetch K and V rows of next iteration's first position (global_prefetch_b8)
        const int sn = s + 16;
        if (sn < L) {
            const int blkn = bt[sn >> 6];
            const size_t bn = (((size_t)blkn * BLK_T + (sn & 63)) * N_HEADS + h) * HEAD_D + lane * 4;
            __builtin_prefetch(Kbase + bn, 0, 0);
            __builtin_prefetch(Vbase + bn, 0, 0);
        }

        float p1 = q.x * k1.x + q.y * k1.y + q.z * k1.z + q.w * k1.w;
        float p2 = q.x * k2.x + q.y * k2.y + q.z * k2.z + q.w * k2.w;
#pragma unroll
        for (int o = 16; o > 0; o >>= 1) {    // two independent butterflies
            p1 += __shfl_xor(p1, o, 32);
            p2 += __shfl_xor(p2, o, 32);
        }
        p1 *= scale;
        p2 = has2 ? p2 * scale : -1e30f;      // tail: weight underflows to 0

        const v4f v1 = __builtin_nontemporal_load((const v4f*)(Vbase + b1));
        v4f vv2 = {0.f, 0.f, 0.f, 0.f};
        if (has2) vv2 = __builtin_nontemporal_load((const v4f*)(Vbase + b2));

        const float nm   = fmaxf(m, fmaxf(p1, p2));
        const float corr = __expf(m - nm);
        const float w1   = __expf(p1 - nm);
        const float w2   = __expf(p2 - nm);
        acc = acc * corr + v1 * w1 + vv2 * w2;
        l   = l * corr + w1 + w2;
        m   = nm;
    }

    __shared__ float sm[8], sl[8], sacc[8][HEAD_D];
    sacc[wave][lane * 4 + 0] = acc[0];
    sacc[wave][lane * 4 + 1] = acc[1];
    sacc[wave][lane * 4 + 2] = acc[2];
    sacc[wave][lane * 4 + 3] = acc[3];
    if (lane == 0) { sm[wave] = m; sl[wave] = l; }
    __syncthreads();

    const int t = threadIdx.x;
    if (t < HEAD_D) {
        float M = sm[0];
#pragma unroll
        for (int w = 1; w < 8; ++w) M = fmaxf(M, sm[w]);
        float lt = 0.f, a = 0.f;
#pragma unroll
        for (int w = 0; w < 8; ++w) {
            const float f = __expf(sm[w] - M);
            lt += sl[w] * f;
            a  += sacc[w][t] * f;
        }
        out[(size_t)b * H_DIM + h * HEAD_D + t] = a / lt;
    }
}

// ---------------------------------------------------------------- launcher
extern "C" void kernel_launch(void* const* d_in, const int* in_sizes, int n_in,
                              void* d_out, int out_size, void* d_ws, size_t ws_size,
                              hipStream_t stream) {
    (void)in_sizes; (void)n_in; (void)out_size; (void)ws_size;

    const float* hidden = (const float*)d_in[0];
    const float* kv     = (const float*)d_in[1];
    const int*   bt     = (const int*)  d_in[2];
    const int*   sl     = (const int*)  d_in[3];
    // d_in[4] = max_seq_len (unused, constant 2048)
    const float* ln1_g  = (const float*)d_in[5];
    const float* ln1_b  = (const float*)d_in[6];
    const float* ln2_g  = (const float*)d_in[7];
    const float* ln2_b  = (const float*)d_in[8];
    const float* w_attn = (const float*)d_in[9];
    const float* b_attn = (const float*)d_in[10];
    const float* w_proj = (const float*)d_in[11];
    const float* b_proj = (const float*)d_in[12];
    const float* w_fc   = (const float*)d_in[13];
    const float* b_fc   = (const float*)d_in[14];
    const float* w_fc2  = (const float*)d_in[15];
    const float* b_fc2  = (const float*)d_in[16];

    float* ws   = (float*)d_ws;
    float* x_ln = ws;                               // 32*2048
    float* qkv  = x_ln + B_SZ * H_DIM;              // 32*6144
    float* attn = qkv  + B_SZ * 3 * H_DIM;          // 32*2048
    float* h    = attn + B_SZ * H_DIM;              // 32*2048
    float* x2   = h    + B_SZ * H_DIM;              // 32*2048
    float* ff   = x2   + B_SZ * H_DIM;              // 32*8192
    float* out  = (float*)d_out;

    ln_kernel<<<B_SZ, 256, 0, stream>>>(hidden, ln1_g, ln1_b, x_ln);

    gemm32<3 * H_DIM, H_DIM, 0, false><<<(3 * H_DIM) / 16, 256, 0, stream>>>(
        x_ln, w_attn, b_attn, nullptr, qkv);

    attn_kernel<<<B_SZ * N_HEADS, 256, 0, stream>>>(qkv, kv, bt, sl, attn);

    gemm32<H_DIM, H_DIM, 0, true><<<H_DIM / 16, 256, 0, stream>>>(
        attn, w_proj, b_proj, hidden, h);

    ln_kernel<<<B_SZ, 256, 0, stream>>>(h, ln2_g, ln2_b, x2);

    gemm32<D_FF, H_DIM, 1, false><<<D_FF / 16, 256, 0, stream>>>(
        x2, w_fc, b_fc, nullptr, ff);

    gemm32<H_DIM, D_FF, 0, true><<<H_DIM / 16, 256, 0, stream>>>(
        ff, w_fc2, b_fc2, h, out);
}